// QModel_65481071399898
// MI455X (gfx1250) — compile-verified
//
#include <hip/hip_runtime.h>

// ---------------------------------------------------------------------------
// 16-qubit real-amplitude circuit simulator for MI455X (gfx1250).
// One workgroup (1024 threads = 32 wave32) per batch element; the full 2^16
// state (256 KB) lives in LDS (CDNA5: 320 KB/WGP). Each RY layer is applied
// as four 16x16 tensor-product matrices via chained V_WMMA_F32_16X16X4_F32.
// CNOT rings are linear maps over GF(2) on basis indices and are *deferred*
// into the LDS addressing (XOR column masks) -- zero data movement for CNOTs.
// Stage-in uses the Tensor Data Mover (tensor_load_to_lds, TENSORcnt):
// 4 descriptors, one 64 KB quarter per wave 0..3, overlapped with gate-matrix
// and LUT construction. Readout folds rho^3 + <Z_i> signs + head weights into
// two 256-entry LUTs and a single block reduction.
// HBM traffic = one pass over the input (128 MB @ 23.3 TB/s ~ 5.5 us floor).
// ---------------------------------------------------------------------------

typedef __attribute__((ext_vector_type(2))) float v2f;
typedef __attribute__((ext_vector_type(8))) float v8f;
typedef __attribute__((ext_vector_type(4))) unsigned int u32x4;
typedef __attribute__((ext_vector_type(8))) int i32x8;
typedef __attribute__((ext_vector_type(4))) int i32x4;

#define NWIRES 16
#define QDIM   65536
#define QDEPTH 3
#define BLOCK  1024

// LDS layout (in floats)
#define OFF_S    0
#define OFF_M    QDIM                        // 12 * 256 group matrices
#define OFF_RED  (QDIM + 3072)
#define OFF_WLO  (QDIM + 3072 + 1024)
#define OFF_WHI  (QDIM + 3072 + 1024 + 256)
#define LDS_FLOATS (QDIM + 3072 + 1024 + 512)

#if __has_builtin(__builtin_amdgcn_tensor_load_to_lds) && \
    __has_builtin(__builtin_amdgcn_s_wait_tensorcnt)
#define USE_TDM 1
#else
#define USE_TDM 0
#endif

// CNOT ring as an index map: gates CNOT(i, i+1) for i=0..14, then CNOT(15,0).
// wire w <-> bit (15-w). Linear over GF(2).
__device__ __forceinline__ unsigned ring_fwd(unsigned p) {
#pragma unroll
    for (int i = 0; i < 15; ++i)
        p ^= ((p >> (15 - i)) & 1u) << (14 - i);
    p ^= (p & 1u) << 15;
    return p;
}

__device__ __forceinline__ unsigned ring_inv(unsigned p) {
    p ^= (p & 1u) << 15;
#pragma unroll
    for (int i = 14; i >= 0; --i)
        p ^= ((p >> (15 - i)) & 1u) << (14 - i);
    return p;
}

__global__ __launch_bounds__(BLOCK, 1) void qsim_kernel(
    const float* __restrict__ state, const float* __restrict__ weights,
    const float* __restrict__ head_w, const float* __restrict__ head_b,
    float* __restrict__ out)
{
    extern __shared__ float lds[];
    float* Sh   = lds + OFF_S;
    float* Mall = lds + OFF_M;
    float* red  = lds + OFF_RED;
    float* wlo  = lds + OFF_WLO;
    float* whi  = lds + OFF_WHI;

    const int tid  = threadIdx.x;
    const int lane = tid & 31;
    const int m16  = lane & 15;   // A row / B,D column index within tile
    const int kh   = lane >> 4;   // lane half: K offset selector
    const int wave = tid >> 5;
    const int b    = blockIdx.x;

    // ---- stage state[b] into LDS ------------------------------------------
#if USE_TDM
    // Tensor Data Mover: 4 descriptors, one contiguous 64 KB quarter per wave.
    // D# group0: count=1, lds_addr[63:32], global_addr[120:64], type=2[127:126]
    // D# group1: data_size=3 (8B), tensor_dim0 = tile_dim0 = 8192 elements,
    //            tensor_dim0_stride = 8192; no multicast/pad/iterate.
    if (wave < 4) {
        const unsigned long long ga =
            (unsigned long long)(const void*)(state + (size_t)b * QDIM) +
            (unsigned long long)wave * 65536ull;
        const unsigned lbase =
            (unsigned)(unsigned long long)(uintptr_t)Sh + (unsigned)wave * 65536u;
        u32x4 g0 = { 1u,                                  // count=1, user mode
                     lbase,                               // lds_addr (bytes)
                     (unsigned)(ga & 0xFFFFFFFFull),      // global_addr[31:0]
                     (unsigned)((ga >> 32) & 0x01FFFFFFull) | 0x80000000u };
        i32x8 g1 = { (int)0x00030000,                     // data_size = 8 bytes
                     (int)(8192u << 16),                  // tensor_dim0 = 8192
                     0,
                     (int)(8192u << 16),                  // tile_dim0 = 8192
                     0,
                     8192,                                // tensor_dim0_stride
                     0, 0 };
        i32x4 gz4 = { 0, 0, 0, 0 };
        i32x8 gz8 = { 0, 0, 0, 0, 0, 0, 0, 0 };
        __builtin_amdgcn_tensor_load_to_lds(g0, g1, gz4, gz4, gz8, 0);
    }
#else
    {
        const float4* gp = (const float4*)(state + (size_t)b * QDIM);
        float4* sp = (float4*)Sh;
#pragma unroll
        for (int j = 0; j < QDIM / 4 / BLOCK; ++j)
            sp[tid + BLOCK * j] = gp[tid + BLOCK * j];
    }
#endif

    // ---- build the 12 (layer,group) 16x16 tensor-product RY matrices -------
    // (overlaps with the TDM transfer)
    if (tid < 256) {
        const int mj = tid >> 4, mk = tid & 15;
#pragma unroll
        for (int dg = 0; dg < 12; ++dg) {
            const int dd = dg >> 2, gg = dg & 3;
            float v = 1.0f;
#pragma unroll
            for (int t = 0; t < 4; ++t) {
                const float th = 0.5f * weights[dd * NWIRES + gg * 4 + t];
                const float c = __cosf(th), sn = __sinf(th);
                const int jb = (mj >> (3 - t)) & 1;
                const int kb = (mk >> (3 - t)) & 1;
                v *= (jb == kb) ? c : (jb ? sn : -sn);
            }
            Mall[dg * 256 + tid] = v;
        }
    } else if (tid < 512) {
        // readout LUTs: w(q) = wlo[q & 255] + whi[q >> 8]
        // q bit bb (bb<8) <-> wire 15-bb ; q bit bb+8 <-> wire 7-bb
        const unsigned x = (unsigned)(tid - 256);
        float alo = 0.0f, ahi = 0.0f;
#pragma unroll
        for (int bb = 0; bb < 8; ++bb) {
            const float sgn = ((x >> bb) & 1u) ? -1.0f : 1.0f;
            alo += sgn * head_w[15 - bb];
            ahi += sgn * head_w[7 - bb];
        }
        wlo[x] = alo;
        whi[x] = ahi;
    }

#if USE_TDM
    if (wave < 4) __builtin_amdgcn_s_wait_tensorcnt(0);
#endif
    __syncthreads();

    // Phi_d columns: physical address of logical unit bit after d deferred rings
    unsigned col[16];
#pragma unroll
    for (int bb = 0; bb < 16; ++bb) col[bb] = 1u << bb;

#pragma unroll
    for (int d = 0; d < QDEPTH; ++d) {
#pragma unroll
        for (int g = 0; g < 4; ++g) {
            const int s = 12 - 4 * g;                 // nibble position of group
            const float* Mg = Mall + (d * 4 + g) * 256;

            // per-lane invariant XOR terms (all col[] indices compile-time)
            unsigned mterm = 0, nterm = 0;
#pragma unroll
            for (int t = 0; t < 4; ++t) {
                if ((m16 >> t) & 1) {
                    mterm ^= col[t < s ? t : t + 4];  // A-row low nibble of r
                    nterm ^= col[s + t];              // D output column nibble
                }
            }
            const unsigned c0    = col[s];
            const unsigned hterm = kh ? col[s + 1] : 0u;
            const unsigned kh3   = kh ? col[3 < s ? 3 : 7] : 0u;
            const unsigned rl0   = col[0 < s ? 0 : 4];
            const unsigned rl1   = col[1 < s ? 1 : 5];
            const unsigned rl2   = col[2 < s ? 2 : 6];

            // B fragment = Mg^T K-slices; shared by all 8 tiles of this wave.
            // Layout: VGPR0 lanes0-15 K=4q, lanes16-31 K=4q+2; VGPR1 K=+1.
            v2f Bf[4];
#pragma unroll
            for (int q = 0; q < 4; ++q) {
                const int kk = 4 * q + 2 * kh;
                Bf[q].x = Mg[m16 * 16 + kk];
                Bf[q].y = Mg[m16 * 16 + kk + 1];
            }

            // 256 tiles of 16 logical rows; 32 waves -> 8 tiles each (disjoint,
            // in-place safe; barrier only between groups)
            for (int tt = 0; tt < 8; ++tt) {
                const unsigned tile = (unsigned)(wave + 32 * tt);
                unsigned baseR0 = 0;
#pragma unroll
                for (int bb = 0; bb < 8; ++bb)
                    if ((tile >> bb) & 1u)
                        baseR0 ^= col[(bb + 4) < s ? (bb + 4) : (bb + 8)];

                // gather ALL K-slice A fragments first so the ds_loads
                // pipeline ahead of the WMMA chain (no per-WMMA dscnt drain)
                const unsigned abase = baseR0 ^ mterm ^ hterm;
                v2f Af[4];
#pragma unroll
                for (int q = 0; q < 4; ++q) {
                    unsigned qt = 0;
                    if (q & 1) qt ^= col[s + 2];
                    if (q & 2) qt ^= col[s + 3];
                    const unsigned a0 = abase ^ qt;
                    Af[q].x = Sh[a0];
                    Af[q].y = Sh[a0 ^ c0];
                }

                v8f acc = {};
#pragma unroll
                for (int q = 0; q < 4; ++q)
                    acc = __builtin_amdgcn_wmma_f32_16x16x4_f32(
                        false, Af[q], false, Bf[q], (short)0, acc, false, false);

                // D: VGPR v -> row (v + 8*kh), column m16
                const unsigned dbase = baseR0 ^ nterm ^ kh3;
#pragma unroll
                for (int v = 0; v < 8; ++v) {
                    unsigned rt = dbase;
                    if (v & 1) rt ^= rl0;
                    if (v & 2) rt ^= rl1;
                    if (v & 4) rt ^= rl2;
                    Sh[rt] = acc[v];
                }
            }
            __syncthreads();
        }

        // Defer this layer's CNOT ring: Phi_{d+1}(e_b) = Phi_d(ring_inv(e_b))
        if (d < QDEPTH - 1) {
            unsigned nc[16];
#pragma unroll
            for (int bb = 0; bb < 16; ++bb) {
                const unsigned m = ring_inv(1u << bb);
                unsigned a = 0;
#pragma unroll
                for (int c = 0; c < 16; ++c)
                    if ((m >> c) & 1u) a ^= col[c];
                nc[bb] = a;
            }
#pragma unroll
            for (int bb = 0; bb < 16; ++bb) col[bb] = nc[bb];
        }
    }

    // ---- readout: logical q = rho^3(p); out[b] = hb + sum p^2 * w(q) -------
    unsigned r1c[16], r2c[16], r3c[16];
#pragma unroll
    for (int bb = 0; bb < 16; ++bb) r1c[bb] = ring_fwd(1u << bb);
#pragma unroll
    for (int bb = 0; bb < 16; ++bb) {
        unsigned a = 0, m = r1c[bb];
#pragma unroll
        for (int c = 0; c < 16; ++c) if ((m >> c) & 1u) a ^= r1c[c];
        r2c[bb] = a;
    }
#pragma unroll
    for (int bb = 0; bb < 16; ++bb) {
        unsigned a = 0, m = r2c[bb];
#pragma unroll
        for (int c = 0; c < 16; ++c) if ((m >> c) & 1u) a ^= r1c[c];
        r3c[bb] = a;
    }

    unsigned qbase = 0;
#pragma unroll
    for (int bb = 0; bb < 10; ++bb)
        if (((unsigned)tid >> bb) & 1u) qbase ^= r3c[bb];

    float partial = 0.0f;
#pragma unroll
    for (int j = 0; j < QDIM / BLOCK; ++j) {
        unsigned jt = 0;
#pragma unroll
        for (int bb = 10; bb < 16; ++bb)
            if (((unsigned)(j << 10) >> bb) & 1u) jt ^= r3c[bb];
        const unsigned q = qbase ^ jt;
        const float a = Sh[tid + BLOCK * j];
        partial += a * a * (wlo[q & 255u] + whi[q >> 8]);
    }

    red[tid] = partial;
    __syncthreads();
#pragma unroll
    for (int sd = BLOCK / 2; sd > 0; sd >>= 1) {
        if (tid < sd) red[tid] += red[tid + sd];
        __syncthreads();
    }
    if (tid == 0) out[b] = red[0] + head_b[0];
}

extern "C" void kernel_launch(void* const* d_in, const int* in_sizes, int n_in,
                              void* d_out, int out_size, void* d_ws, size_t ws_size,
                              hipStream_t stream) {
    (void)n_in; (void)out_size; (void)d_ws; (void)ws_size;
    const float* state   = (const float*)d_in[0];
    const float* weights = (const float*)d_in[1];
    const float* head_w  = (const float*)d_in[2];
    const float* head_b  = (const float*)d_in[3];
    float* out = (float*)d_out;

    const int batch = in_sizes[0] / QDIM;
    const size_t ldsB = (size_t)LDS_FLOATS * sizeof(float);  // ~274 KB < 320 KB/WGP

    (void)hipFuncSetAttribute((const void*)qsim_kernel,
                              hipFuncAttributeMaxDynamicSharedMemorySize,
                              (int)ldsB);
    qsim_kernel<<<dim3(batch), dim3(BLOCK), ldsB, stream>>>(
        state, weights, head_w, head_b, out);
}